// AttentionFlow_12627203850790
// MI455X (gfx1250) — compile-verified
//
#include <hip/hip_runtime.h>
#include <hip/hip_bf16.h>

typedef __attribute__((ext_vector_type(2))) float v2f;
typedef __attribute__((ext_vector_type(8))) float v8f;

#define Bc 32
#define Tt 1024
#define Jj 128
#define Dd 256
#define QPITCH 260   // 260 % 64 = 4 -> conflict-free column access; 1040B row pitch (16B multiple)
#define CPITCH 260
#define SPITCH 132   // 132 % 64 = 4

// Kernel 1: per (batch, 16-row t-tile). 256 threads = 8 waves.
// Computes sim tile (16t x 128j) via WMMA, row softmax, attended_query tile
// (16t x 256d) via WMMA, writes output segments 0..2 and per-row sim-max to ws.
__global__ __launch_bounds__(256)
void af_k1(const float* __restrict__ ctx, const float* __restrict__ qry,
           const float* __restrict__ w, const float* __restrict__ bptr,
           float* __restrict__ out, float* __restrict__ rowmax_g)
{
    __shared__ __attribute__((aligned(16))) float qs[Jj * QPITCH];   // query[b], [j][d], padded
    __shared__ __attribute__((aligned(16))) float cms[16 * CPITCH];  // context tile * w_m
    __shared__ __attribute__((aligned(16))) float sims[16 * SPITCH]; // sim, then attn, [t][j]
    __shared__ float scs[16];
    __shared__ float sqs[Jj];

    const int tid  = threadIdx.x;
    const int lane = tid & 31;
    const int wave = tid >> 5;
    const int b    = blockIdx.x >> 6;          // T/16 = 64 tiles per batch
    const int t0   = (blockIdx.x & 63) << 4;

    const float bias = bptr[0];
    const float* ctxb = ctx + ((size_t)b * Tt + t0) * Dd;
    const float* qryb = qry + (size_t)b * Jj * Dd;

    // ---- Async global->LDS staging of query[b] (CDNA5 ASYNCcnt path). ----
    // Each lane moves 16B; one wave covers 128 floats (512B); each row (1024B)
    // takes two issues sharing one address pair via the immediate offset
    // (inst_offset is added to BOTH the LDS and global addresses).
    {
        const int lane4 = lane << 2;
        const int jbeg  = wave << 4;           // 16 rows per wave
        for (int j = jbeg; j < jbeg + 16; ++j) {
            const float*  ga = qryb + j * Dd + lane4;
            const unsigned la = (unsigned)(unsigned long long)(&qs[j * QPITCH + lane4]);
            asm volatile(
                "global_load_async_to_lds_b128 %0, %1, off\n\t"
                "global_load_async_to_lds_b128 %0, %1, off offset:512"
                :: "v"(la), "v"(ga) : "memory");
        }
    }

    // ---- Stage context tile scaled by w_m (float4 path, overlaps async). ----
    {
        const int c4 = (tid & 63) << 2;        // column group: 0,4,...,252
        const int r0 = tid >> 6;               // 0..3
        const float4 wm4 = *(const float4*)&w[2 * Dd + c4];
        for (int r = r0; r < 16; r += 4) {
            float4 v = *(const float4*)&ctxb[r * Dd + c4];
            float4 o;
            o.x = v.x * wm4.x; o.y = v.y * wm4.y;
            o.z = v.z * wm4.z; o.w = v.w * wm4.w;
            *(float4*)&cms[r * CPITCH + c4] = o;
        }
    }
    // s_c[t] = context[b,t,:] . w_c  (16 threads per row; float4 dots)
    {
        const int r = tid >> 4, l = tid & 15;
        const int d0 = l << 4;
        float p = 0.f;
        #pragma unroll
        for (int i = 0; i < 16; i += 4) {
            float4 cv = *(const float4*)&ctxb[r * Dd + d0 + i];
            float4 wv = *(const float4*)&w[d0 + i];
            p += cv.x * wv.x + cv.y * wv.y + cv.z * wv.z + cv.w * wv.w;
        }
        p += __shfl_xor(p, 1, 32);
        p += __shfl_xor(p, 2, 32);
        p += __shfl_xor(p, 4, 32);
        p += __shfl_xor(p, 8, 32);
        if (l == 0) scs[r] = p;
    }

    asm volatile("s_wait_asynccnt 0x0" ::: "memory");  // query resident in LDS
    __syncthreads();

    // s_q[j] = query[b,j,:] . w_q   (2 threads per j, halves of D, float4 dots)
    {
        const int j = tid >> 1, half = tid & 1;
        const int d0 = half * 128;
        float p = 0.f;
        for (int d = d0; d < d0 + 128; d += 4) {
            float4 qv = *(const float4*)&qs[j * QPITCH + d];
            float4 wv = *(const float4*)&w[Dd + d];
            p += qv.x * wv.x + qv.y * wv.y + qv.z * wv.z + qv.w * wv.w;
        }
        p += __shfl_xor(p, 1, 32);
        if (half == 0) sqs[j] = p;
    }
    __syncthreads();

    // GEMM 1: simT (16j x 16t) per wave, A = query rows, B = (ctx*w_m)^T, K = D.
    {
        const int jr = (wave << 4) + (lane & 15);  // A: M = j
        const int tt = lane & 15;                   // B: N = t
        const int kk = (lane >> 4) << 1;
        const int hi = (lane >> 4) << 3;
        const float* qrow = &qs[jr * QPITCH];
        const float* crow = &cms[tt * CPITCH];
        v8f c = {};
        for (int k = 0; k < Dd; k += 4) {
            v2f a, bm;
            a.x  = qrow[k + kk];
            a.y  = qrow[k + kk + 1];
            bm.x = crow[k + kk];
            bm.y = crow[k + kk + 1];
            c = __builtin_amdgcn_wmma_f32_16x16x4_f32(
                    false, a, false, bm, (short)0, c, false, false);
        }
        // D layout: element v -> M = j = 16*wave + v + hi, N = t = lane&15.
        #pragma unroll
        for (int v = 0; v < 8; ++v) {
            const int j = (wave << 4) + v + hi;
            sims[tt * SPITCH + j] = c[v] + scs[tt] + sqs[j] + bias;
        }
    }
    __syncthreads();

    // Row softmax over j (16 threads per row t); write row-max for beta pass.
    {
        const int t = tid >> 4, l = tid & 15;
        float e[8];
        float m = -3.4e38f;
        #pragma unroll
        for (int i = 0; i < 8; ++i) {
            e[i] = sims[t * SPITCH + l + 16 * i];
            m = fmaxf(m, e[i]);
        }
        m = fmaxf(m, __shfl_xor(m, 1, 32));
        m = fmaxf(m, __shfl_xor(m, 2, 32));
        m = fmaxf(m, __shfl_xor(m, 4, 32));
        m = fmaxf(m, __shfl_xor(m, 8, 32));
        if (l == 0) rowmax_g[(size_t)b * Tt + t0 + t] = m;
        float s = 0.f;
        #pragma unroll
        for (int i = 0; i < 8; ++i) { e[i] = __expf(e[i] - m); s += e[i]; }
        s += __shfl_xor(s, 1, 32);
        s += __shfl_xor(s, 2, 32);
        s += __shfl_xor(s, 4, 32);
        s += __shfl_xor(s, 8, 32);
        const float inv = 1.0f / s;
        #pragma unroll
        for (int i = 0; i < 8; ++i)
            sims[t * SPITCH + l + 16 * i] = e[i] * inv;
    }
    __syncthreads();

    // GEMM 2: attended (16t x 256d), A = attn (M=t, K=j), B = query (K=j, N=d).
    {
        const int tt = lane & 15;                  // A: M = t
        const int kk = (lane >> 4) << 1;
        const int hi = (lane >> 4) << 3;
        for (int dt = wave; dt < 16; dt += 8) {    // 2 d-tiles per wave
            const int d = (dt << 4) + (lane & 15); // B: N = d
            v8f c = {};
            for (int k = 0; k < Jj; k += 4) {
                v2f a, bm;
                a.x  = sims[tt * SPITCH + k + kk];
                a.y  = sims[tt * SPITCH + k + kk + 1];
                bm.x = qs[(k + kk) * QPITCH + d];
                bm.y = qs[(k + kk + 1) * QPITCH + d];
                c = __builtin_amdgcn_wmma_f32_16x16x4_f32(
                        false, a, false, bm, (short)0, c, false, false);
            }
            #pragma unroll
            for (int v = 0; v < 8; ++v) {
                const int t = v + hi;
                const float aq = c[v];
                const float cv = ctxb[t * Dd + d];
                const size_t base = ((size_t)(b * Tt + t0 + t)) * (4 * Dd);
                out[base + d]          = cv;        // segment 0: context
                out[base + Dd + d]     = aq;        // segment 1: attended_query
                out[base + 2 * Dd + d] = cv * aq;   // segment 2: ctx * aq
            }
        }
    }
}

// Kernel 2: one block per batch. beta = softmax_t(rowmax); attended_context.
__global__ __launch_bounds__(256)
void af_k2(const float* __restrict__ ctx, const float* __restrict__ rowmax_g,
           float* __restrict__ ac)
{
    __shared__ float beta[Tt];
    __shared__ float red[8];
    const int b = blockIdx.x, tid = threadIdx.x;
    const int lane = tid & 31, wave = tid >> 5;

    float v[4];
    float m = -3.4e38f;
    #pragma unroll
    for (int i = 0; i < 4; ++i) {
        v[i] = rowmax_g[(size_t)b * Tt + tid + 256 * i];
        m = fmaxf(m, v[i]);
    }
    for (int o = 1; o < 32; o <<= 1) m = fmaxf(m, __shfl_xor(m, o, 32));
    if (lane == 0) red[wave] = m;
    __syncthreads();
    m = red[0];
    for (int i = 1; i < 8; ++i) m = fmaxf(m, red[i]);
    float s = 0.f;
    #pragma unroll
    for (int i = 0; i < 4; ++i) { v[i] = __expf(v[i] - m); s += v[i]; }
    for (int o = 1; o < 32; o <<= 1) s += __shfl_xor(s, o, 32);
    __syncthreads();
    if (lane == 0) red[wave] = s;
    __syncthreads();
    s = 0.f;
    for (int i = 0; i < 8; ++i) s += red[i];
    const float inv = 1.0f / s;
    #pragma unroll
    for (int i = 0; i < 4; ++i) beta[tid + 256 * i] = v[i] * inv;
    __syncthreads();

    // attended_context[b, d=tid] = sum_t beta[t] * context[b,t,d]
    float acc = 0.f;
    const float* cb = ctx + (size_t)b * Tt * Dd + tid;
    for (int t = 0; t < Tt; ++t)
        acc += beta[t] * cb[(size_t)t * Dd];
    ac[b * Dd + tid] = acc;
}

// Kernel 3: segment 3 = context * attended_context (broadcast over t).
__global__ __launch_bounds__(256)
void af_k3(const float* __restrict__ ctx, const float* __restrict__ ac,
           float* __restrict__ out)
{
    const size_t idx = (size_t)blockIdx.x * 256 + threadIdx.x;  // B*T*D
    const int d = (int)(idx & (Dd - 1));
    const size_t bt = idx >> 8;          // b*T + t
    const int b = (int)(bt >> 10);
    out[bt * (4 * Dd) + 3 * Dd + d] = ctx[idx] * ac[b * Dd + d];
}

extern "C" void kernel_launch(void* const* d_in, const int* in_sizes, int n_in,
                              void* d_out, int out_size, void* d_ws, size_t ws_size,
                              hipStream_t stream) {
    const float* ctx = (const float*)d_in[0];  // (B,T,D)
    const float* qry = (const float*)d_in[1];  // (B,J,D)
    const float* w   = (const float*)d_in[2];  // (3D,)
    const float* bia = (const float*)d_in[3];  // (1,)
    float* out = (float*)d_out;                // (B,T,4D)

    float* rowmax_g = (float*)d_ws;            // B*T floats  (128 KB)
    float* ac       = rowmax_g + Bc * Tt;      // B*D floats  (32 KB)

    af_k1<<<Bc * (Tt / 16), 256, 0, stream>>>(ctx, qry, w, bia, out, rowmax_g);
    af_k2<<<Bc, 256, 0, stream>>>(ctx, rowmax_g, ac);
    af_k3<<<(Bc * Tt * Dd) / 256, 256, 0, stream>>>(ctx, ac, out);
}